// SignalGenerationLayer_46600395162211
// MI455X (gfx1250) — compile-verified
//
#include <hip/hip_runtime.h>

typedef float v2f __attribute__((ext_vector_type(2)));
typedef float v8f __attribute__((ext_vector_type(8)));

// ---- model constants (double, folded at compile time) ----
#define GAMMA_D 267522100.0
#define B0_D    3.0
#define DCHI_D  2.64e-07
#define HCT_D   0.34
#define TE_D    0.074
#define R2T_D   11.5
#define TR_D    3.0
#define TI_D    0.8
#define T1B_D   1.58
#define NB_D    0.775
#define TD_D    0.0045067
#define PI_D    3.14159265358979323846

// dw = C_DW * oef
constexpr double C_DW_D = (4.0 / 3.0) * PI_D * GAMMA_D * B0_D * DCHI_D * HCT_D;
// K = C_K * oef^2  (0.5*gamma^2 * g0 * td^2 with g0 = 4/45*hct*(1-hct)*(dchi*B0*oef)^2)
constexpr double C_K_D = 0.5 * GAMMA_D * GAMMA_D * (4.0 / 45.0) * HCT_D * (1.0 - HCT_D) *
                         (DCHI_D * B0_D) * (DCHI_D * B0_D) * TD_D * TD_D;
constexpr double STEP_D   = (1.0 - 1e-5) / 256.0;   // linspace step == h
constexpr double H3_D     = STEP_D / 3.0;           // h/3 Simpson factor
constexpr double INV_TD_D = 1.0 / TD_D;
constexpr double TETD_D   = TE_D / TD_D;

#define N_PTS 257
#define N_PAD 260           // pad to multiple of 4 (v=0 on the pad)
#define K_ITERS 65          // 260 / 4
#define TAU_STEP_F 0.008f
#define TAU_START_F (-0.016f)

// Numerical-Recipes J0, small-|x| rational part (valid |x| < 8).
__device__ __forceinline__ float j0_small(float xin) {
    float y  = xin * xin;
    float tn = fmaf(y, -184.9052456f, 77392.33017f);
    tn = fmaf(y, tn, -11214424.18f);
    tn = fmaf(y, tn, 651619640.7f);
    tn = fmaf(y, tn, -13362590354.0f);
    tn = fmaf(y, tn, 57568490574.0f);
    float td = y + 267.8532712f;
    td = fmaf(y, td, 59272.64853f);
    td = fmaf(y, td, 9494680.718f);
    td = fmaf(y, td, 1029532985.0f);
    td = fmaf(y, td, 57568490411.0f);
    return tn * __builtin_amdgcn_rcpf(td);
}

// Full J0, branchless (both paths, select at end).
__device__ __forceinline__ float j0_full(float xin) {
    float ax    = fabsf(xin);
    float small = j0_small(xin);
    float axl = fmaxf(ax, 8.0f);
    float r   = __builtin_amdgcn_rcpf(axl);
    float z   = 8.0f * r;
    float y2  = z * z;
    float xx  = axl - 0.785398164f;
    float p1 = fmaf(y2, 2.093887211e-7f, -2.073370639e-6f);
    p1 = fmaf(y2, p1, 2.734510407e-5f);
    p1 = fmaf(y2, p1, -1.098628627e-3f);
    p1 = fmaf(y2, p1, 1.0f);
    float p2 = fmaf(y2, -9.34935152e-8f, 7.621095161e-7f);
    p2 = fmaf(y2, p2, -6.911147651e-6f);
    p2 = fmaf(y2, p2, 1.430488765e-4f);
    p2 = fmaf(y2, p2, -1.562499995e-2f);
    float amp   = __builtin_amdgcn_sqrtf(0.636619772f * r);
    float large = amp * (__cosf(xx) * p1 - z * __sinf(xx) * p2);
    return (ax < 8.0f) ? small : large;
}

__global__ __launch_bounds__(256)
void qbold_wmma_kernel(const float* __restrict__ x, float* __restrict__ out,
                       int n_vox, int P) {
    // Per-block tables: u_i and combined Simpson weights v_i = s_i*(h/3)*W(u_i)
    __shared__ __align__(16) float s_u[N_PAD];
    __shared__ __align__(16) float s_v[N_PAD];
    for (int i = threadIdx.x; i < N_PAD; i += 256) {
        float u = 1.0f, v = 0.0f;
        if (i < N_PTS) {
            float uu = fmaf((float)i, (float)STEP_D, 1e-5f);
            float sw = (i == 0 || i == N_PTS - 1) ? 1.0f : ((i & 1) ? 4.0f : 2.0f);
            float W  = (2.0f + uu) * __builtin_amdgcn_sqrtf(fmaxf(1.0f - uu, 0.0f)) /
                       (3.0f * uu * uu);
            u = uu;
            v = W * sw * (float)H3_D;
        }
        s_u[i] = u;
        s_v[i] = v;
    }
    __syncthreads();

    const int lane = threadIdx.x & 31;
    const int wave = threadIdx.x >> 5;
    const int tile = blockIdx.x * 8 + wave;
    const int row  = lane & 15;                 // A-matrix row this lane serves

    int p = tile * 16 + row;                    // (voxel,tau) pair index
    const bool active = (p < P);
    if (p >= P) p = P - 1;                      // clamp: keep EXEC all-ones for WMMA
    const int   voxel = p / 11;
    const int   t     = p - voxel * 11;
    const float tau   = fmaf((float)t, TAU_STEP_F, TAU_START_F);
    const float oef   = x[2 * voxel];
    const float dbv   = x[2 * voxel + 1];
    const float carg  = 1.5f * tau * ((float)C_DW_D * oef);   // arg = carg * u

    // Wave-uniform split point: chunks whose max arg stays < 8 for ALL lanes
    // need only the cheap rational J0 (no sin/cos/sqrt asymptotic path).
    float m = fabsf(carg);
    m = fmaxf(m, __shfl_xor(m, 16, 32));
    m = fmaxf(m, __shfl_xor(m, 8, 32));
    m = fmaxf(m, __shfl_xor(m, 4, 32));
    m = fmaxf(m, __shfl_xor(m, 2, 32));
    m = fmaxf(m, __shfl_xor(m, 1, 32));
    // all-small for u-index i with m*u_i < 8  ->  i < lim; chunk c covers 4c..4c+3
    const float lim = (8.0f / fmaxf(m, 1e-30f) - 1e-5f) * (float)(1.0 / STEP_D);
    int c_split = (int)fminf(fmaxf((lim - 3.0f) * 0.25f - 1.0f, 0.0f), (float)K_ITERS);
    // Pin to SGPR: value is wave-uniform by construction; this lets the compiler
    // use scalar loop latches (no EXEC masking), structurally guaranteeing the
    // WMMA "EXEC must be all ones" requirement and removing per-iter exec ops.
    c_split = __builtin_amdgcn_readfirstlane(c_split);

    // wave32 WMMA 16x16x4 f32: lanes 0-15 hold K={0,1}, lanes 16-31 K={2,3}
    const int klo = (lane & 16) ? 2 : 0;

    v8f acc = {0.f, 0.f, 0.f, 0.f, 0.f, 0.f, 0.f, 0.f};
    int c = 0;
    // Phase 1: all args < 8 -> rational path only (no transcendentals but rcp)
    for (; c < c_split; ++c) {
        const int k = c * 4 + klo;
        const float2 up = *(const float2*)(s_u + k);
        const float2 vp = *(const float2*)(s_v + k);
        v2f a, b;
        a.x = 1.0f - j0_small(carg * up.x);
        a.y = 1.0f - j0_small(carg * up.y);
        b.x = vp.x;
        b.y = vp.y;
        acc = __builtin_amdgcn_wmma_f32_16x16x4_f32(
            false, a, false, b, (short)0, acc, false, false);
    }
    // Phase 2: mixed magnitudes -> full branchless J0
    for (; c < K_ITERS; ++c) {
        const int k = c * 4 + klo;
        const float2 up = *(const float2*)(s_u + k);
        const float2 vp = *(const float2*)(s_v + k);
        v2f a, b;
        a.x = 1.0f - j0_full(carg * up.x);
        a.y = 1.0f - j0_full(carg * up.y);
        b.x = vp.x;
        b.y = vp.y;
        acc = __builtin_amdgcn_wmma_f32_16x16x4_f32(
            false, a, false, b, (short)0, acc, false, false);
    }

    // Extract integ for row `row`. D layout: VGPR j, lanes 0-15 -> D[j, n=lane],
    // lanes 16-31 -> D[8+j, n=lane-16]. Row l<8: own acc[l] (column l == lane l).
    // Row l>=8: lane l+8 holds D[8+(l-8), l-8] in acc[l-8].
    const int rsel = lane & 7;
    float sel = acc[0];
    sel = (rsel == 1) ? acc[1] : sel;
    sel = (rsel == 2) ? acc[2] : sel;
    sel = (rsel == 3) ? acc[3] : sel;
    sel = (rsel == 4) ? acc[4] : sel;
    sel = (rsel == 5) ? acc[5] : sel;
    sel = (rsel == 6) ? acc[6] : sel;
    sel = (rsel == 7) ? acc[7] : sel;
    const int src = (lane < 8) ? lane : (lane + 8);
    const float integ = __shfl(sel, src, 32);

    if (lane < 16 && active) {
        // tissue
        const float tissue = __expf(fmaf(-dbv, integ, -(float)(TE_D * R2T_D)));
        // blood
        const float oef2 = oef * oef;
        const float r2b  = (float)(4.5 + 16.4 * HCT_D) +
                           (float)(165.2 * HCT_D + 55.7) * oef2;
        const float K    = (float)C_K_D * oef2;
        const float tp   = (float)TE_D + tau;
        const float tm   = (float)TE_D - tau;
        const float S    = (float)TETD_D +
                           __builtin_amdgcn_sqrtf(0.25f + (float)TETD_D) + 1.5f -
                           2.0f * __builtin_amdgcn_sqrtf(fmaf(tp * tp, (float)INV_TD_D, 0.25f)) -
                           2.0f * __builtin_amdgcn_sqrtf(fmaf(tm * tm, (float)INV_TD_D, 0.25f));
        const float blood = __expf(fmaf(-r2b, (float)TE_D, -K * S));
        // mixing
        const float e1 = __expf((float)(-(TR_D - TI_D) / T1B_D));
        const float e2 = __expf((float)(-TI_D / T1B_D));
        const float m_bld = 1.0f - (2.0f - e1) * e2;
        const float bw = m_bld * (float)NB_D * dbv;
        out[p] = fmaf(-bw, tissue, tissue) + bw * blood;
    }
}

extern "C" void kernel_launch(void* const* d_in, const int* in_sizes, int n_in,
                              void* d_out, int out_size, void* d_ws, size_t ws_size,
                              hipStream_t stream) {
    const float* x   = (const float*)d_in[0];
    float*       out = (float*)d_out;
    const int n_vox  = in_sizes[0] / 2;           // (oef,dbv) pairs
    const long long P = (long long)n_vox * 11;    // total (voxel,tau) outputs
    const long long tiles  = (P + 15) / 16;       // 16 outputs per wave-tile
    const long long blocks = (tiles + 7) / 8;     // 8 waves (tiles) per 256-thread block
    qbold_wmma_kernel<<<(int)blocks, 256, 0, stream>>>(x, out, n_vox, (int)P);
}